// DSConv_62569083568818
// MI455X (gfx1250) — compile-verified
//
#include <hip/hip_runtime.h>
#include <math.h>

typedef float v2f __attribute__((ext_vector_type(2)));
typedef float v8f __attribute__((ext_vector_type(8)));

#define BB 2
#define CIN 128
#define COUT 128
#define WW 128
#define HH 128
#define KK 9
#define MM (WW * HH)            // 16384 pixels per batch
#define KDIM (CIN * KK)         // 1152
#define ASTRIDE 1156            // padded LDS row stride (stride%64==4 -> conflict-free b64)
#define TILEM 32                // pixels per workgroup (2 WMMA M-tiles)
#define GN_GROUPS 32
#define EPS 1e-5f

// ---------------- workspace layout (float offsets; all 16B aligned) ----------------
#define OFF_FT    0u                       // fT   [B*M, C]          4,194,304 f
#define OFF_BT    4194304u                 // Bt   [KDIM, COUT]        147,456 f
#define OFF_IDX   4341760u                 // int4 [B*M*K]           1,179,648 i
#define OFF_W4    5521408u                 // f4   [B*M*K]           1,179,648 f
#define OFF_CONV  6701056u                 // conv [B*M, COUT]       4,194,304 f
#define OFF_SUM   10895360u                // [B*32]
#define OFF_SSQ   10895424u                // [B*32]

// ================= kernel 1: f [B,C,P] -> fT [B,P,C] tiled transpose =================
__global__ __launch_bounds__(256) void transpose_kernel(const float* __restrict__ f,
                                                        float* __restrict__ fT) {
    __shared__ float tile[32][33];
    int b = blockIdx.z;
    int p0 = blockIdx.x * 32, c0 = blockIdx.y * 32;
    int tx = threadIdx.x, ty = threadIdx.y;   // 32 x 8
    const float* src = f + (size_t)b * CIN * MM;
    float* dst = fT + (size_t)b * MM * CIN;
#pragma unroll
    for (int j = 0; j < 32; j += 8)
        tile[ty + j][tx] = src[(size_t)(c0 + ty + j) * MM + p0 + tx];
    __syncthreads();
#pragma unroll
    for (int j = 0; j < 32; j += 8)
        dst[(size_t)(p0 + ty + j) * CIN + c0 + tx] = tile[tx][ty + j];
}

// ================= kernel 2: conv_w [Cout,Cin,1,K] -> Bt [k*128+c][co] =================
__global__ __launch_bounds__(256) void bt_kernel(const float* __restrict__ conv_w,
                                                 float* __restrict__ Bt) {
    int e = blockIdx.x * 256 + threadIdx.x;
    if (e >= KDIM * COUT) return;
    int kk = e >> 7, co = e & 127;
    int k = kk >> 7, c = kk & 127;
    Bt[e] = conv_w[(size_t)co * KDIM + c * KK + k];
}

// ==== kernel 3: offset conv3x3 (ch 9..17) + BN + tanh + cumsum + bilinear setup ====
__global__ __launch_bounds__(256) void offset_sample_kernel(
    const float* __restrict__ f, const float* __restrict__ off_w,
    const float* __restrict__ off_b, const float* __restrict__ bn_gamma,
    const float* __restrict__ bn_beta, const float* __restrict__ bn_mean,
    const float* __restrict__ bn_var, int4* __restrict__ samp_idx,
    float4* __restrict__ samp_w) {
    __shared__ float wS[KK * CIN * 9];  // wS[k*1152 + c*9 + t], 10368 f = 41.5 KB
    for (int i = threadIdx.x; i < KK * CIN * 9; i += 256) {
        int k = i / (CIN * 9);
        int rem = i - k * (CIN * 9);
        wS[i] = off_w[(size_t)(KK + k) * (CIN * 9) + rem];  // x_offset = channels 9..17
    }
    __syncthreads();

    int gid = blockIdx.x * 256 + threadIdx.x;   // over B*M
    int b = gid / MM;
    int m = gid - b * MM;
    int w = m >> 7, h = m & 127;

    float acc[KK];
#pragma unroll
    for (int k = 0; k < KK; ++k) acc[k] = 0.f;

    for (int c = 0; c < CIN; ++c) {
        const float* fb = f + (((size_t)(b * CIN + c)) << 14);
        float fv[9];
#pragma unroll
        for (int dy = 0; dy < 3; ++dy) {
#pragma unroll
            for (int dx = 0; dx < 3; ++dx) {
                int yy = w + dy - 1, xx = h + dx - 1;
                bool ok = (yy >= 0) && (yy < WW) && (xx >= 0) && (xx < HH);
                fv[dy * 3 + dx] = ok ? fb[yy * HH + xx] : 0.f;
            }
        }
#pragma unroll
        for (int k = 0; k < KK; ++k) {
#pragma unroll
            for (int t = 0; t < 9; ++t)
                acc[k] = fmaf(fv[t], wS[k * (CIN * 9) + c * 9 + t], acc[k]);
        }
    }

    float p[KK];
#pragma unroll
    for (int k = 0; k < KK; ++k) {
        int ch = KK + k;
        float v = acc[k] + off_b[ch];
        v = (v - bn_mean[ch]) * rsqrtf(bn_var[ch] + EPS);
        v = v * bn_gamma[ch] + bn_beta[ch];
        p[k] = tanhf(v);
    }
    // cumulative from center (CENTER=4)
    float nw[KK];
    nw[0] = p[0]; nw[8] = p[8]; nw[4] = 0.f;
    nw[5] = p[5]; nw[6] = nw[5] + p[6]; nw[7] = nw[6] + p[7];
    nw[3] = p[3]; nw[2] = nw[3] + p[2]; nw[1] = nw[2] + p[1];

    int base = b * MM;
#pragma unroll
    for (int k = 0; k < KK; ++k) {
        float y = (float)w + (float)(k - 4);
        float x = (float)h + nw[k];
        int y0r = (int)floorf(y), x0r = (int)floorf(x);
        int y1 = min(max(y0r + 1, 0), WW - 1);
        int x1 = min(max(x0r + 1, 0), HH - 1);
        int y0 = min(max(y0r, 0), WW - 1);
        int x0 = min(max(x0r, 0), HH - 1);
        float y0f = (float)y0, y1f = (float)y1, x0f = (float)x0, x1f = (float)x1;
        float wa0 = (y1f - y) * (x1f - x);
        float wc0 = (y1f - y) * (x - x0f);
        float wa1 = (y - y0f) * (x1f - x);
        float wc1 = (y - y0f) * (x - x0f);
        int s = gid * KK + k;
        samp_idx[s] = make_int4(base + y0 * HH + x0, base + y0 * HH + x1,
                                base + y1 * HH + x0, base + y1 * HH + x1);
        samp_w[s] = make_float4(wa0, wc0, wa1, wc1);
    }
}

// ================= kernel 4: zero GN accumulators =================
__global__ void init_stats_kernel(float* __restrict__ gsum, float* __restrict__ gssq) {
    int i = threadIdx.x;
    if (i < BB * GN_GROUPS) { gsum[i] = 0.f; gssq[i] = 0.f; }
}

// ===== kernel 5: fused bilinear gather + snake GEMM (WMMA f32 16x16x4) + bias + stats =====
// 32-pixel M tile: two accumulators per wave share each B fragment (halves B traffic).
__global__ __launch_bounds__(256) void snake_gemm_kernel(
    const float* __restrict__ fT, const float* __restrict__ Bt,
    const int4* __restrict__ samp_idx, const float4* __restrict__ samp_w,
    const float* __restrict__ conv_b, float* __restrict__ conv_out,
    float* __restrict__ gsum, float* __restrict__ gssq) {
    __shared__ float As[TILEM * ASTRIDE];   // ~148 KB (2 WGs / WGP on 320KB LDS)

    int b = blockIdx.x >> 9;             // 512 M-tiles per batch
    int mt = blockIdx.x & 511;
    int m0 = mt * TILEM;
    int tid = threadIdx.x;

    // ---- phase A: build A tile [32 pixels x 1152] via 4-tap gather ----
    for (int e = tid; e < TILEM * KDIM; e += 256) {
        int pr = e / KDIM;              // pixel row in tile
        int r = e - pr * KDIM;          // kk = k*128 + c
        int k = r >> 7, c = r & 127;
        int s = ((b * MM + m0 + pr) * KK) + k;
        int4 i4 = samp_idx[s];
        float4 w4 = samp_w[s];
        float val = w4.x * fT[(size_t)i4.x * CIN + c] +
                    w4.y * fT[(size_t)i4.y * CIN + c] +
                    w4.z * fT[(size_t)i4.z * CIN + c] +
                    w4.w * fT[(size_t)i4.w * CIN + c];
        As[pr * ASTRIDE + r] = val;
    }
    __syncthreads();

    // ---- phase B: 8 waves x (32x16 out tile), K loop with v_wmma_f32_16x16x4_f32 ----
    int lane = tid & 31;
    int wv = tid >> 5;                   // 0..7 -> N tile
    int n0 = wv * 16;
    int ln = lane & 15;
    int co = n0 + ln;
    int khi = (lane & 16) >> 3;          // 0 for lanes 0-15, 2 for lanes 16-31

    v8f acc0 = {}, acc1 = {};
    const float* arow0 = &As[ln * ASTRIDE];
    const float* arow1 = &As[(ln + 16) * ASTRIDE];
    for (int kk = 0; kk < KDIM; kk += 4) {
        int klo = kk + khi;
        v2f a0 = *(const v2f*)(arow0 + klo);               // A[M=ln   ][klo, klo+1]
        v2f a1 = *(const v2f*)(arow1 + klo);               // A[M=ln+16][klo, klo+1]
        v2f bf;
        bf.x = Bt[(size_t)klo * COUT + co];                // B[klo][N=ln]
        bf.y = Bt[(size_t)(klo + 1) * COUT + co];
        acc0 = __builtin_amdgcn_wmma_f32_16x16x4_f32(
            false, a0, false, bf, (short)0, acc0, false, false);
        acc1 = __builtin_amdgcn_wmma_f32_16x16x4_f32(
            false, a1, false, bf, (short)0, acc1, false, false);
    }

    // ---- bias, store [B, M, Cout], GN partial stats ----
    float bias = conv_b[co];
    int mBase0 = m0 + ((lane < 16) ? 0 : 8);
    int mBase1 = mBase0 + 16;
    float s = 0.f, s2 = 0.f;
#pragma unroll
    for (int r = 0; r < 8; ++r) {
        float v = acc0[r] + bias;
        conv_out[(size_t)(b * MM + mBase0 + r) * COUT + co] = v;
        s += v; s2 += v * v;
    }
#pragma unroll
    for (int r = 0; r < 8; ++r) {
        float v = acc1[r] + bias;
        conv_out[(size_t)(b * MM + mBase1 + r) * COUT + co] = v;
        s += v; s2 += v * v;
    }
    int g = b * GN_GROUPS + (co >> 2);
    atomicAdd(&gsum[g], s);
    atomicAdd(&gssq[g], s2);
}

// ================= kernel 6: GroupNorm + ReLU, write d_out [B,Cout,W,H] =================
__global__ __launch_bounds__(256) void gn_apply_kernel(
    const float* __restrict__ conv_out, const float* __restrict__ gsum,
    const float* __restrict__ gssq, const float* __restrict__ gn_gamma,
    const float* __restrict__ gn_beta, float* __restrict__ out) {
    int e = blockIdx.x * 256 + threadIdx.x;     // = (b*COUT + co)*M + m
    if (e >= BB * COUT * MM) return;
    int m = e & (MM - 1);
    int bc = e >> 14;
    int co = bc & 127;
    int b = bc >> 7;
    float cnt = 4.0f * (float)MM;
    int g = b * GN_GROUPS + (co >> 2);
    float mean = gsum[g] / cnt;
    float var = gssq[g] / cnt - mean * mean;
    float inv = rsqrtf(var + EPS);
    float v = conv_out[(size_t)(b * MM + m) * COUT + co];
    v = (v - mean) * inv * gn_gamma[co] + gn_beta[co];
    out[e] = v > 0.f ? v : 0.f;
}

extern "C" void kernel_launch(void* const* d_in, const int* in_sizes, int n_in,
                              void* d_out, int out_size, void* d_ws, size_t ws_size,
                              hipStream_t stream) {
    const float* f        = (const float*)d_in[0];
    const float* off_w    = (const float*)d_in[1];
    const float* off_b    = (const float*)d_in[2];
    const float* bn_gamma = (const float*)d_in[3];
    const float* bn_beta  = (const float*)d_in[4];
    const float* bn_mean  = (const float*)d_in[5];
    const float* bn_var   = (const float*)d_in[6];
    const float* conv_w   = (const float*)d_in[7];
    const float* conv_b   = (const float*)d_in[8];
    const float* gn_gamma = (const float*)d_in[9];
    const float* gn_beta  = (const float*)d_in[10];
    float* out = (float*)d_out;

    float* ws = (float*)d_ws;
    float* fT       = ws + OFF_FT;
    float* Bt       = ws + OFF_BT;
    int4*  samp_idx = (int4*)(ws + OFF_IDX);
    float4* samp_w  = (float4*)(ws + OFF_W4);
    float* conv_out = ws + OFF_CONV;
    float* gsum     = ws + OFF_SUM;
    float* gssq     = ws + OFF_SSQ;

    transpose_kernel<<<dim3(MM / 32, CIN / 32, BB), dim3(32, 8), 0, stream>>>(f, fT);
    bt_kernel<<<(KDIM * COUT + 255) / 256, 256, 0, stream>>>(conv_w, Bt);
    offset_sample_kernel<<<(BB * MM) / 256, 256, 0, stream>>>(
        f, off_w, off_b, bn_gamma, bn_beta, bn_mean, bn_var, samp_idx, samp_w);
    init_stats_kernel<<<1, 128, 0, stream>>>(gsum, gssq);
    snake_gemm_kernel<<<BB * (MM / TILEM), 256, 0, stream>>>(
        fT, Bt, samp_idx, samp_w, conv_b, conv_out, gsum, gssq);
    gn_apply_kernel<<<(BB * COUT * MM + 255) / 256, 256, 0, stream>>>(
        conv_out, gsum, gssq, gn_gamma, gn_beta, out);
}